// Fully_Connected_GAT_Processor_82042465288300
// MI455X (gfx1250) — compile-verified
//
#include <hip/hip_runtime.h>
#include <hip/hip_bf16.h>
#include <math.h>

// MI455X / gfx1250, wave32. fp32 WMMA 16x16x4 for all true GEMMs; the
// leaky-inside-reduction attention-score pass stays on VALU (not factorable).
// j-tiles of g are streamed via GLOBAL_LOAD_ASYNC_TO_LDS_B128 (ASYNCcnt),
// double-buffered so the copy hides behind the VALU e-phase.

typedef __attribute__((ext_vector_type(2))) float v2f;
typedef __attribute__((ext_vector_type(8))) float v8f;

#define N_NODES 2048
#define HDIM    64
#define NHEADS  4
#define NHID    16
#define LDIM    16
#define INFEAT  80
#define TJ      128
#define NT      (N_NODES / TJ)
#define SLOPE   0.2f

__device__ __forceinline__ float leaky(float v) { return v >= 0.0f ? v : SLOPE * v; }

// Issue one 32KB tile (TJ x HDIM f32) as 16 async b128 loads per lane.
// p in [0,128): lane id among the four loader waves. LDS dest address is the
// low 32 bits of the generic shared pointer (LDS aperture: addr[31:0]).
__device__ __forceinline__ void issue_gj_tile(const float* __restrict__ g, int t,
                                              const float* lds_dst, int p) {
    uint32_t lbase = (uint32_t)(uintptr_t)lds_dst;
    uint64_t gbase = (uint64_t)(uintptr_t)(g + (size_t)t * TJ * HDIM);
#pragma unroll
    for (int r = 0; r < 16; ++r) {
        uint32_t elem = (uint32_t)(r * 128 + p) * 16u;   // 16-byte units
        uint32_t la = lbase + elem;
        uint64_t ga = gbase + elem;
        asm volatile("global_load_async_to_lds_b128 %0, %1, off"
                     :: "v"(la), "v"(ga) : "memory");
    }
}

__device__ __forceinline__ void wait_async0() {
    asm volatile("s_wait_asynccnt 0" ::: "memory");
}

// ---------------- Kernel 0: fill unlabeled ohm rows with 1/L ----------------
__global__ void k_fill_ohm(const float* __restrict__ ohm_labels, float* __restrict__ ohm_f) {
    int i = blockIdx.x * blockDim.x + threadIdx.x;
    if (i >= N_NODES) return;
    float v[LDIM];
    float s = 0.0f;
#pragma unroll
    for (int l = 0; l < LDIM; ++l) { v[l] = ohm_labels[i * LDIM + l]; s += v[l]; }
    bool unl = (s == 0.0f);
#pragma unroll
    for (int l = 0; l < LDIM; ++l) ohm_f[i * LDIM + l] = unl ? (1.0f / LDIM) : v[l];
}

// ---------------- Kernel 1: g = [x|ohm] @ W_l^T  (2048x80 @ 80x64) ----------
__global__ void k_proj(const float* __restrict__ x, const float* __restrict__ ohm_f,
                       const float* __restrict__ W_l, float* __restrict__ g) {
    __shared__ float pin[16][INFEAT];
    const int i0 = blockIdx.x * 16;
    const int tid = threadIdx.x;             // 128 threads
    for (int idx = tid; idx < 16 * INFEAT; idx += 128) {
        int r = idx / INFEAT, k = idx % INFEAT;
        pin[r][k] = (k < HDIM) ? x[(i0 + r) * HDIM + k]
                               : ohm_f[(i0 + r) * LDIM + (k - HDIM)];
    }
    __syncthreads();
    const int wave = tid >> 5, lane = tid & 31;
    const int c0 = wave * 16;
    const int m = lane & 15;                 // row (A) / col (B,D) within tile
    const int koff = (lane >> 4) * 2;        // lanes 16-31 hold K+2,K+3
    v8f acc = {};
    for (int k0 = 0; k0 < INFEAT; k0 += 4) {
        v2f a, b;
        a.x = pin[m][k0 + koff];
        a.y = pin[m][k0 + koff + 1];
        b.x = W_l[(c0 + m) * INFEAT + k0 + koff];       // B[k][c] = W_l[c][k]
        b.y = W_l[(c0 + m) * INFEAT + k0 + koff + 1];
        acc = __builtin_amdgcn_wmma_f32_16x16x4_f32(false, a, false, b,
                                                    (short)0, acc, false, false);
    }
#pragma unroll
    for (int r = 0; r < 8; ++r) {
        int mi = r + (lane >> 4) * 8;
        g[(i0 + mi) * HDIM + c0 + m] = acc[r];
    }
}

// ---------------- Kernel 2: fused GAT attention (flash style) ----------------
// Block = 16 query rows, 256 threads (8 waves). j-tiles of g are async-copied
// to double-buffered LDS by waves 4-7 (issued before the VALU e-phase, drained
// during the WMMA phase). VALU computes e with leaky inside; online softmax;
// waves 0..3 (one per head) accumulate p @ g_j with V_WMMA_F32_16X16X4_F32.
__global__ void k_attn(const float* __restrict__ g, const float* __restrict__ W_attn,
                       float* __restrict__ hout) {
    __shared__ float gjb[2][TJ][HDIM];           // 64 KB, double-buffered
    __shared__ float pbuf[NHEADS][16][TJ];       // 32 KB (e, then p = exp)
    __shared__ float gi[16][HDIM];               // 4 KB
    __shared__ float watt[NHID];
    __shared__ float red[64][4];
    __shared__ float mrow[NHEADS][16];
    __shared__ float srow[NHEADS][16];
    __shared__ float scl[NHEADS][16];

    const int i0 = blockIdx.x * 16;
    const int tid = threadIdx.x;                 // 256
    const int wave = tid >> 5, lane = tid & 31;
    const int head = wave;                       // valid when wave < 4

    // kick off tile 0 immediately (waves 4-7), overlap with gi staging
    if (wave >= NHEADS) issue_gj_tile(g, 0, &gjb[0][0][0], tid & 127);

    for (int idx = tid; idx < 16 * HDIM; idx += 256)
        gi[idx >> 6][idx & 63] = g[(i0 + (idx >> 6)) * HDIM + (idx & 63)];
    if (tid < NHID) watt[tid] = W_attn[tid];
    if (tid < 64) { mrow[tid >> 4][tid & 15] = -1e30f; srow[tid >> 4][tid & 15] = 0.0f; }

    if (wave >= NHEADS) wait_async0();
    __syncthreads();

    const int pair = tid >> 2;                   // 64 (head,row) pairs
    const int h = pair >> 4, irow = pair & 15;
    const int sub = tid & 3;
    const int jbase = sub * 32;

    // hoist per-thread invariants for the e-phase (constant across all tiles)
    float giv[NHID], wv[NHID];
#pragma unroll
    for (int f = 0; f < NHID; ++f) {
        giv[f] = gi[irow][h * NHID + f];
        wv[f]  = watt[f];
    }

    v8f acc = {};

    for (int t = 0; t < NT; ++t) {
        const int buf = t & 1;
        const float (*gj)[HDIM] = gjb[buf];

        // issue next tile before the long VALU phase (non-blocking)
        if (wave >= NHEADS && t + 1 < NT)
            issue_gj_tile(g, t + 1, &gjb[buf ^ 1][0][0], tid & 127);

        // e[i,j,h] = sum_f w_f * leaky(gi + gj); track running max
        float lmax = -1e30f;
        for (int jj = jbase; jj < jbase + 32; ++jj) {
            float e = 0.0f;
#pragma unroll
            for (int f = 0; f < NHID; ++f) {
                float v = giv[f] + gj[jj][h * NHID + f];
                e = fmaf(leaky(v), wv[f], e);
            }
            pbuf[h][irow][jj] = e;
            lmax = fmaxf(lmax, e);
        }
        red[pair][sub] = lmax;
        __syncthreads();
        if (sub == 0) {
            float mold = mrow[h][irow];
            float mnew = fmaxf(fmaxf(red[pair][0], red[pair][1]),
                               fmaxf(red[pair][2], red[pair][3]));
            mnew = fmaxf(mnew, mold);
            scl[h][irow] = __expf(mold - mnew);
            mrow[h][irow] = mnew;
        }
        __syncthreads();
        const float mnew = mrow[h][irow];
        float lsum = 0.0f;
        for (int jj = jbase; jj < jbase + 32; ++jj) {
            float pe = __expf(pbuf[h][irow][jj] - mnew);
            pbuf[h][irow][jj] = pe;
            lsum += pe;
        }
        red[pair][sub] = lsum;
        __syncthreads();
        if (sub == 0) {
            srow[h][irow] = srow[h][irow] * scl[h][irow]
                          + red[pair][0] + red[pair][1] + red[pair][2] + red[pair][3];
        }
        __syncthreads();

        // Phase B: waves 0-3 run acc = acc*scale + P(16xTJ) @ Gj(TJx16);
        // waves 4-7 drain their async copy of the next tile.
        if (wave < NHEADS) {
            const int m = lane & 15;
            const int koff = (lane >> 4) * 2;
#pragma unroll
            for (int r = 0; r < 8; ++r) {
                int mi = r + (lane >> 4) * 8;
                acc[r] *= scl[head][mi];
            }
            for (int k0 = 0; k0 < TJ; k0 += 4) {
                v2f a, b;
                a.x = pbuf[head][m][k0 + koff];
                a.y = pbuf[head][m][k0 + koff + 1];
                b.x = gj[k0 + koff][head * NHID + m];
                b.y = gj[k0 + koff + 1][head * NHID + m];
                acc = __builtin_amdgcn_wmma_f32_16x16x4_f32(false, a, false, b,
                                                            (short)0, acc, false, false);
            }
        } else if (t + 1 < NT) {
            wait_async0();
        }
        __syncthreads();
    }

    if (wave < NHEADS) {
        const int m = lane & 15;
#pragma unroll
        for (int r = 0; r < 8; ++r) {
            int mi = r + (lane >> 4) * 8;
            float val = acc[r] / srow[head][mi];
            hout[(i0 + mi) * HDIM + head * NHID + m] = leaky(val);
        }
    }
}

// ---------------- Kernel 3: centroids = (ohm^T @ h) / cnt, missing ----------
__global__ void k_centroid(const float* __restrict__ ohm_f, const float* __restrict__ hmat,
                           float* __restrict__ cent, float* __restrict__ missing_out) {
    __shared__ float cnt_part[8][LDIM];
    __shared__ float cnt[LDIM];
    const int tid = threadIdx.x;                 // 256
    const int wave = tid >> 5, lane = tid & 31;

    if (tid >= 128) {
        int l = (tid - 128) & 15;
        int chunk = (tid - 128) >> 4;            // 8 chunks of 256 rows
        float s = 0.0f;
        for (int i = chunk * 256; i < (chunk + 1) * 256; ++i) s += ohm_f[i * LDIM + l];
        cnt_part[chunk][l] = s;
    }

    v8f acc = {};
    if (wave < 4) {
        const int c0 = wave * 16;
        const int m = lane & 15;
        const int koff = (lane >> 4) * 2;
        for (int k0 = 0; k0 < N_NODES; k0 += 4) {
            int ka = k0 + koff;
            v2f a, b;
            a.x = ohm_f[ka * LDIM + m];          // A[l][k] = ohm^T
            a.y = ohm_f[(ka + 1) * LDIM + m];
            b.x = hmat[ka * HDIM + c0 + m];
            b.y = hmat[(ka + 1) * HDIM + c0 + m];
            acc = __builtin_amdgcn_wmma_f32_16x16x4_f32(false, a, false, b,
                                                        (short)0, acc, false, false);
        }
    }
    __syncthreads();
    if (tid < LDIM) {
        float s = 0.0f;
#pragma unroll
        for (int c = 0; c < 8; ++c) s += cnt_part[c][tid];
        cnt[tid] = s;
        missing_out[tid] = (s == 0.0f) ? 1.0f : 0.0f;
    }
    __syncthreads();
    if (wave < 4) {
        const int c0 = wave * 16;
        const int m = lane & 15;
#pragma unroll
        for (int r = 0; r < 8; ++r) {
            int l = r + (lane >> 4) * 8;
            float c = cnt[l];
            cent[l * HDIM + c0 + m] = (c == 0.0f) ? 0.0f : acc[r] / c;
        }
    }
}

// ---------------- Kernel 4: scores = -cdist(h, centroids) -------------------
__global__ void k_scores(const float* __restrict__ hmat, const float* __restrict__ cent,
                         float* __restrict__ scores) {
    __shared__ float cs[LDIM][HDIM];
    const int tid = threadIdx.x;                 // 256
    for (int idx = tid; idx < LDIM * HDIM; idx += 256) cs[idx >> 6][idx & 63] = cent[idx];
    __syncthreads();
    const int i = blockIdx.x * 256 + tid;
    float hrow[HDIM];
#pragma unroll
    for (int d = 0; d < HDIM; ++d) hrow[d] = hmat[i * HDIM + d];
#pragma unroll
    for (int l = 0; l < LDIM; ++l) {
        float s = 0.0f;
#pragma unroll
        for (int d = 0; d < HDIM; ++d) {
            float df = hrow[d] - cs[l][d];
            s = fmaf(df, df, s);
        }
        scores[i * LDIM + l] = -sqrtf(s);
    }
}

extern "C" void kernel_launch(void* const* d_in, const int* in_sizes, int n_in,
                              void* d_out, int out_size, void* d_ws, size_t ws_size,
                              hipStream_t stream) {
    const float* x          = (const float*)d_in[0];
    const float* ohm_labels = (const float*)d_in[1];
    const float* W_l        = (const float*)d_in[2];
    const float* W_attn     = (const float*)d_in[3];
    // d_in[4] = sparsity (unused, eval mode)

    float* out     = (float*)d_out;
    float* scores  = out;                                   // 2048*16
    float* hmat    = out + N_NODES * LDIM;                  // 2048*64
    float* missing = out + N_NODES * LDIM + N_NODES * HDIM; // 16

    float* g     = (float*)d_ws;                            // 2048*64
    float* ohm_f = g + N_NODES * HDIM;                      // 2048*16
    float* cent  = ohm_f + N_NODES * LDIM;                  // 16*64

    k_fill_ohm<<<N_NODES / 256, 256, 0, stream>>>(ohm_labels, ohm_f);
    k_proj    <<<N_NODES / 16, 128, 0, stream>>>(x, ohm_f, W_l, g);
    k_attn    <<<N_NODES / 16, 256, 0, stream>>>(g, W_attn, hmat);
    k_centroid<<<1, 256, 0, stream>>>(ohm_f, hmat, cent, missing);
    k_scores  <<<N_NODES / 256, 256, 0, stream>>>(hmat, cent, scores);
}